// TransformerEncoder_274877907223
// MI455X (gfx1250) — compile-verified
//
#include <hip/hip_runtime.h>
#include <hip/hip_bf16.h>
#include <math.h>

// ---------------------------------------------------------------------------
// MI455X (gfx1250, wave32) implementation of the VN-Performer encoder block.
// All GEMMs run on V_WMMA_F32_16X16X4_F32 (exact fp32 MAC) — the problem is
// compute-bound (~97 GFLOP vs ~45us of HBM traffic at 23.3 TB/s), and fp32
// WMMA preserves the reference's fp32 numerics feeding exp()/layernorm.
// ---------------------------------------------------------------------------

typedef float v2f __attribute__((ext_vector_type(2)));
typedef float v8f __attribute__((ext_vector_type(8)));

// Shapes (fixed by the reference)
#define Bn   4
#define Cc   256
#define Nn   4096
#define Hh   8
#define Jj   (3 * Nn)          // 12288 columns per batch in channel-mix GEMMs
#define BHn  (Bn * Hh)         // 32
#define Mf   256               // random-feature count m
#define HIDn 512

#define DN    0.31947225f      // 96^-0.25
#define DNSQ  0.10206207f      // 96^-0.5
#define RATIO 0.0625f          // 256^-0.5
#define VEPS  1e-6f
#define LNEPS 1e-5f
#define KEPS  1e-4f

__device__ __forceinline__ v8f wmma4(v2f a, v2f b, v8f c) {
  // D(16x16,f32) = A(16x4,f32) x B(4x16,f32) + C
  return __builtin_amdgcn_wmma_f32_16x16x4_f32(false, a, false, b, (short)0, c,
                                               false, false);
}

// ---------------------------------------------------------------------------
// Channel-mixing GEMM: y[b,o,v,n] = sum_c W[o,c] * x[b,c,v,n]
// As GEMM: Y_b (O x J) = W (O x Cin) x X_b (Cin x J), J = 3N, all row-major.
// Block = 8 waves; tile = 64 (o) x 128 (j). W tile staged in LDS (pitch 130
// -> conflict-free ds_load_b64 for the A operand pairs).
// ---------------------------------------------------------------------------
__global__ void cmix_gemm(const float* __restrict__ X, const float* __restrict__ W,
                          float* __restrict__ Y, int O, int Cin) {
  extern __shared__ float sW[];                 // [64][130]
  const int lane = threadIdx.x & 31, wv = threadIdx.x >> 5;
  const int b = blockIdx.z;
  const int row0 = blockIdx.y * 64;             // o base
  const int col0 = blockIdx.x * 128 + wv * 16;  // j base (per wave)
  const int mr = lane & 15;
  const int k2 = (lane >> 4) * 2;
  const float* Xb = X + (size_t)b * Cin * Jj;
  float* Yb = Y + (size_t)b * O * Jj;

  v8f acc[4];
  for (int t = 0; t < 4; ++t)
    for (int g = 0; g < 8; ++g) acc[t][g] = 0.0f;

  for (int kc = 0; kc < Cin; kc += 128) {
    __syncthreads();
    for (int e = threadIdx.x; e < 64 * 128; e += 256) {
      int r = e >> 7, c = e & 127;
      sW[r * 130 + c] = W[(size_t)(row0 + r) * Cin + kc + c];
    }
    __syncthreads();
    for (int kb = 0; kb < 128; kb += 4) {
      v2f bf;
      bf.x = Xb[(size_t)(kc + kb + k2) * Jj + col0 + mr];
      bf.y = Xb[(size_t)(kc + kb + k2 + 1) * Jj + col0 + mr];
#pragma unroll
      for (int t = 0; t < 4; ++t) {
        v2f af = *(const v2f*)&sW[(t * 16 + mr) * 130 + kb + k2];
        acc[t] = wmma4(af, bf, acc[t]);
      }
    }
  }
#pragma unroll
  for (int t = 0; t < 4; ++t)
#pragma unroll
    for (int g = 0; g < 8; ++g)
      Yb[(size_t)(row0 + t * 16 + g + (lane >> 4) * 8) * Jj + col0 + mr] = acc[t][g];
}

// ---------------------------------------------------------------------------
// diag[b,h,n] = 0.5 * dn^2 * sum_d feat(b,h,n,d)^2  (d = cph*3+v, 96 values)
// lin layout (B,C,3,N): offset for (cph,v) from head base is cv*N.
// ---------------------------------------------------------------------------
__global__ void diag_kernel(const float* __restrict__ lin, float* __restrict__ dg) {
  const int bh = blockIdx.x, b = bh >> 3, h = bh & 7;
  const int n = blockIdx.y * 256 + threadIdx.x;
  const float* p = lin + ((size_t)(b * Cc + h * 32) * 3) * Nn + n;
  float s = 0.f;
  for (int cv = 0; cv < 96; ++cv) {
    float v = p[(size_t)cv * Nn];
    s += v * v;
  }
  dg[bh * Nn + n] = 0.5f * s * DNSQ;
}

// ---------------------------------------------------------------------------
// Feature GEMM: dd[b,h,n,m] = dn * sum_d feat(n,d) * proj[m,d].
// M-dim = n, N-dim = m, K = d (96). proj chunk transposed into LDS (B operand
// is a conflict-free LDS read; A operand is a lane-coalesced global gather).
// Each wave holds 16 n-rows x full 256 m-cols (16 v8f accs) so the query-side
// rowmax + exp epilogue fuses in-registers (shfl_xor within wave32 halves).
// Key side stores raw dn-scaled dd (global stabilizer applied later).
// ---------------------------------------------------------------------------
__global__ void feat_gemm(const float* __restrict__ lin, const float* __restrict__ proj,
                          const float* __restrict__ diag, float* __restrict__ outf,
                          int is_query) {
  extern __shared__ float sproj[];              // [32][256] chunk of proj^T
  const int lane = threadIdx.x & 31, wv = threadIdx.x >> 5;
  const int bh = blockIdx.y, b = bh >> 3, h = bh & 7;
  const int row0 = blockIdx.x * 128 + wv * 16;  // n base for this wave
  const int mr = lane & 15, half = lane >> 4;
  const int k2 = half * 2;
  const int n = row0 + mr;                      // A row for this lane
  const size_t headbase = ((size_t)(b * Cc + h * 32) * 3) * Nn + n;

  v8f acc[16];
  for (int t = 0; t < 16; ++t)
    for (int g = 0; g < 8; ++g) acc[t][g] = 0.0f;

  for (int kc = 0; kc < 96; kc += 32) {
    __syncthreads();
    for (int e = threadIdx.x; e < 32 * 256; e += 256) {
      int dd = e >> 8, mm = e & 255;
      sproj[dd * 256 + mm] = proj[(size_t)mm * 96 + kc + dd];
    }
    __syncthreads();
    for (int kb = 0; kb < 32; kb += 4) {
      const int d0 = kc + kb + k2;
      v2f a;
      a.x = lin[headbase + (size_t)d0 * Nn];          // feat(n, d0)
      a.y = lin[headbase + (size_t)(d0 + 1) * Nn];    // feat(n, d0+1)
#pragma unroll
      for (int t = 0; t < 16; ++t) {
        v2f bf;
        bf.x = sproj[(kb + k2) * 256 + t * 16 + mr];
        bf.y = sproj[(kb + k2 + 1) * 256 + t * 16 + mr];
        acc[t] = wmma4(a, bf, acc[t]);
      }
    }
  }

  if (is_query) {
#pragma unroll
    for (int g = 0; g < 8; ++g) {
      float mx = acc[0][g];
#pragma unroll
      for (int t = 1; t < 16; ++t) mx = fmaxf(mx, acc[t][g]);
      for (int off = 1; off < 16; off <<= 1) mx = fmaxf(mx, __shfl_xor(mx, off, 32));
      const int nr = row0 + g + half * 8;
      const float sub = mx * DN + diag[bh * Nn + nr];
#pragma unroll
      for (int t = 0; t < 16; ++t) {
        float v = acc[t][g] * DN - sub;
        outf[((size_t)bh * Nn + nr) * Mf + t * 16 + mr] = RATIO * (expf(v) + KEPS);
      }
    }
  } else {
#pragma unroll
    for (int g = 0; g < 8; ++g) {
      const int nr = row0 + g + half * 8;
#pragma unroll
      for (int t = 0; t < 16; ++t)
        outf[((size_t)bh * Nn + nr) * Mf + t * 16 + mr] = acc[t][g] * DN;
    }
  }
}

// --------------------------- global max over dd_k ---------------------------
__global__ void gmax_part(const float* __restrict__ d, float* __restrict__ part) {
  __shared__ float sm[256];
  const long chunk = 16384;
  long s0 = (long)blockIdx.x * chunk;
  float mx = -3.0e38f;
  for (long i = s0 + threadIdx.x; i < s0 + chunk; i += 256) mx = fmaxf(mx, d[i]);
  sm[threadIdx.x] = mx;
  __syncthreads();
  for (int off = 128; off; off >>= 1) {
    if (threadIdx.x < off) sm[threadIdx.x] = fmaxf(sm[threadIdx.x], sm[threadIdx.x + off]);
    __syncthreads();
  }
  if (threadIdx.x == 0) part[blockIdx.x] = sm[0];
}

__global__ void gmax_fin(const float* __restrict__ part, float* __restrict__ outv) {
  __shared__ float sm[256];
  float mx = -3.0e38f;
  for (int i = threadIdx.x; i < 2048; i += 256) mx = fmaxf(mx, part[i]);
  sm[threadIdx.x] = mx;
  __syncthreads();
  for (int off = 128; off; off >>= 1) {
    if (threadIdx.x < off) sm[threadIdx.x] = fmaxf(sm[threadIdx.x], sm[threadIdx.x + off]);
    __syncthreads();
  }
  if (threadIdx.x == 0) outv[0] = sm[0];
}

// kf[b,h,n,m] = ratio * (exp(dd - diag_k[n] - kstab) + KEPS), in place
__global__ void finish_kf(float* __restrict__ kf, const float* __restrict__ dg,
                          const float* __restrict__ stab) {
  const long idx = (long)blockIdx.x * 256 + threadIdx.x;
  const int m = (int)(idx & 255);
  (void)m;
  const long r = idx >> 8;
  const int n = (int)(r & (Nn - 1));
  const int bh = (int)(r >> 12);
  float v = kf[idx];
  kf[idx] = RATIO * (expf(v - dg[bh * Nn + n] - stab[0]) + KEPS);
}

// ksum[b,h,m] = sum_n kf[b,h,n,m]
__global__ void ksum_kernel(const float* __restrict__ kf, float* __restrict__ ks) {
  const int idx = blockIdx.x * 256 + threadIdx.x;   // bh*256+m
  const int m = idx & 255, bh = idx >> 8;
  const float* p = kf + (size_t)bh * Nn * Mf + m;
  float s = 0.f;
  for (int n = 0; n < Nn; ++n) s += p[(size_t)n * Mf];
  ks[idx] = s;
}

// vperm[b,h,n,cv] = vlin[b, h*32 + cv/3, cv%3, n]
__global__ void vperm_kernel(const float* __restrict__ vlin, float* __restrict__ vp) {
  const int idx = blockIdx.x * 256 + threadIdx.x;   // over BH*N*96
  const int cv = idx % 96;
  const int rest = idx / 96;
  const int n = rest & (Nn - 1);
  const int bh = rest >> 12, b = bh >> 3, h = bh & 7;
  vp[idx] = vlin[((size_t)(b * Cc + h * 32) * 3 + cv) * Nn + n];
}

// ctx[b,h,m,cv] = sum_n kf[b,h,n,m] * vperm[b,h,n,cv]   (one wave per tile)
__global__ void ctx_gemm(const float* __restrict__ kf, const float* __restrict__ vp,
                         float* __restrict__ ctx) {
  const int lane = threadIdx.x;
  const int bh = blockIdx.z;
  const int row0 = blockIdx.y * 16;  // m rows
  const int col0 = blockIdx.x * 16;  // cv cols
  const int mr = lane & 15, k2 = (lane >> 4) * 2;
  const float* kb = kf + (size_t)bh * Nn * Mf;
  const float* vb = vp + (size_t)bh * Nn * 96;
  v8f acc;
  for (int g = 0; g < 8; ++g) acc[g] = 0.0f;
  for (int k = 0; k < Nn; k += 4) {
    v2f a, bf;
    a.x = kb[(size_t)(k + k2) * Mf + row0 + mr];
    a.y = kb[(size_t)(k + k2 + 1) * Mf + row0 + mr];
    bf.x = vb[(size_t)(k + k2) * 96 + col0 + mr];
    bf.y = vb[(size_t)(k + k2 + 1) * 96 + col0 + mr];
    acc = wmma4(a, bf, acc);
  }
#pragma unroll
  for (int g = 0; g < 8; ++g)
    ctx[((size_t)bh * Mf + row0 + g + (lane >> 4) * 8) * 96 + col0 + mr] = acc[g];
}

// d_inv[b,h,n] = 1 / sum_m qf[b,h,n,m] * ksum[b,h,m]   (one wave per n)
__global__ void dinv_kernel(const float* __restrict__ qf, const float* __restrict__ ks,
                            float* __restrict__ di) {
  const int lane = threadIdx.x & 31, wv = threadIdx.x >> 5;
  const int gn = blockIdx.x * 8 + wv;  // bh*N + n
  const int bh = gn >> 12, n = gn & (Nn - 1);
  const float* q = qf + ((size_t)bh * Nn + n) * Mf;
  const float* k = ks + bh * Mf;
  float s = 0.f;
  for (int m = lane; m < Mf; m += 32) s += q[m] * k[m];
  for (int off = 16; off; off >>= 1) s += __shfl_xor(s, off, 32);
  if (lane == 0) di[gn] = 1.f / s;
}

// attn[b, cph*H + h, v, n] = d_inv[n] * sum_m qf[n,m] * ctx[m, cph*3+v]
__global__ void out_gemm(const float* __restrict__ qf, const float* __restrict__ ctx,
                         const float* __restrict__ di, float* __restrict__ attn) {
  const int lane = threadIdx.x & 31, wv = threadIdx.x >> 5;
  const int bh = blockIdx.z, b = bh >> 3, h = bh & 7;
  const int row0 = (blockIdx.y * 8 + wv) * 16;  // n
  const int col0 = blockIdx.x * 16;             // cv
  const int mr = lane & 15, k2 = (lane >> 4) * 2;
  const float* qb = qf + (size_t)bh * Nn * Mf;
  const float* cb = ctx + (size_t)bh * Mf * 96;
  v8f acc;
  for (int g = 0; g < 8; ++g) acc[g] = 0.0f;
  for (int k = 0; k < Mf; k += 4) {
    v2f a = *(const v2f*)&qb[(size_t)(row0 + mr) * Mf + k + k2];
    v2f bf;
    bf.x = cb[(k + k2) * 96 + col0 + mr];
    bf.y = cb[(k + k2 + 1) * 96 + col0 + mr];
    acc = wmma4(a, bf, acc);
  }
  const int cv = col0 + mr, cph = cv / 3, v = cv % 3;
#pragma unroll
  for (int g = 0; g < 8; ++g) {
    const int n = row0 + g + (lane >> 4) * 8;
    attn[((size_t)(b * Cc + cph * Hh + h) * 3 + v) * Nn + n] = acc[g] * di[bh * Nn + n];
  }
}

// dst = vn_layernorm(src; gamma,beta) + skip    (block = one (b,n), thread = c)
__global__ void ln_res(const float* __restrict__ src, const float* __restrict__ skip,
                       const float* __restrict__ gamma, const float* __restrict__ beta,
                       float* __restrict__ dst) {
  __shared__ float s1[256], s2[256];
  const int n = blockIdx.x, b = blockIdx.y, c = threadIdx.x;
  const size_t base = ((size_t)(b * Cc + c) * 3) * Nn + n;
  const float x0 = src[base], x1 = src[base + Nn], x2 = src[base + 2 * Nn];
  const float nr = sqrtf(x0 * x0 + x1 * x1 + x2 * x2) + VEPS;
  s1[c] = nr;
  s2[c] = nr * nr;
  __syncthreads();
  for (int off = 128; off; off >>= 1) {
    if (c < off) { s1[c] += s1[c + off]; s2[c] += s2[c + off]; }
    __syncthreads();
  }
  const float mu = s1[0] * (1.f / 256.f);
  const float var = s2[0] * (1.f / 256.f) - mu * mu;
  const float ln = (nr - mu) * rsqrtf(var + LNEPS) * gamma[c] + beta[c];
  const float sc = ln / nr;
  dst[base] = x0 * sc + skip[base];
  dst[base + Nn] = x1 * sc + skip[base + Nn];
  dst[base + 2 * Nn] = x2 * sc + skip[base + 2 * Nn];
}

// vn_leaky_relu in place over h (d precomputed): r = x - 0.8*(dot<0 ? dot/(dsq+eps) : 0)*d
__global__ void leaky_kernel(const float* __restrict__ dbuf, float* __restrict__ h) {
  const int idx = blockIdx.x * 256 + threadIdx.x;  // over B*HID*N
  const int n = idx & (Nn - 1);
  const int bo = idx >> 12;
  const size_t base = ((size_t)bo * 3) * Nn + n;
  const float x0 = h[base], x1 = h[base + Nn], x2 = h[base + 2 * Nn];
  const float d0 = dbuf[base], d1 = dbuf[base + Nn], d2 = dbuf[base + 2 * Nn];
  const float dot = x0 * d0 + x1 * d1 + x2 * d2;
  const float dsq = d0 * d0 + d1 * d1 + d2 * d2;
  const float m = (dot >= 0.f) ? 0.f : (dot / (dsq + VEPS));
  h[base] = x0 - 0.8f * m * d0;
  h[base + Nn] = x1 - 0.8f * m * d1;
  h[base + 2 * Nn] = x2 - 0.8f * m * d2;
}

// ---------------------------------------------------------------------------
extern "C" void kernel_launch(void* const* d_in, const int* in_sizes, int n_in,
                              void* d_out, int out_size, void* d_ws, size_t ws_size,
                              hipStream_t stream) {
  const float* x      = (const float*)d_in[0];
  const float* w_proj = (const float*)d_in[1];
  const float* Wq     = (const float*)d_in[2];
  const float* Wk     = (const float*)d_in[3];
  const float* Wv     = (const float*)d_in[4];
  const float* g_attn = (const float*)d_in[5];
  const float* b_attn = (const float*)d_in[6];
  const float* W_mlp1 = (const float*)d_in[7];
  const float* W_dir  = (const float*)d_in[8];
  const float* W_mlp2 = (const float*)d_in[9];
  const float* g_mlp  = (const float*)d_in[10];
  const float* b_mlp  = (const float*)d_in[11];
  float* out = (float*)d_out;
  float* ws = (float*)d_ws;

  const size_t LIN = (size_t)Bn * Cc * 3 * Nn;        // 12,582,912 floats
  const size_t FEAT = (size_t)BHn * Nn * Mf;          // 33,554,432 floats

  // workspace layout with buffer reuse (~475 MB peak)
  float* qlin = ws;                 float* attn = qlin;   // qlin dead after feat_gemm
  float* klin = ws + LIN;           float* skip = klin;   // klin dead after feat_gemm
  float* vlin = ws + 2 * LIN;       float* mlp2 = vlin;   // vlin dead after vperm
  float* kf   = ws + 3 * LIN;       float* h1   = kf;     // kf dead after ctx_gemm
  float* qf   = ws + 3 * LIN + FEAT; float* dbuf = qf;    // qf dead after out_gemm
  float* vp   = ws + 3 * LIN + 2 * FEAT;
  float* smx  = ws + 3 * LIN + 2 * FEAT + LIN;
  float* diagq = smx;
  float* diagk = smx + 131072;
  float* ksumb = smx + 262144;
  float* dinvb = smx + 270336;
  float* ctxb  = smx + 401408;
  float* partb = smx + 1187840;
  float* kstab = smx + 1189888;

  const dim3 blk(256);
  const size_t shW = 64 * 130 * sizeof(float);    // cmix LDS (33,280 B)
  const size_t shP = 32 * 256 * sizeof(float);    // feat LDS (32,768 B)

  // 1) q/k/v channel-mixing GEMMs
  cmix_gemm<<<dim3(Jj / 128, Cc / 64, Bn), blk, shW, stream>>>(x, Wq, qlin, Cc, Cc);
  cmix_gemm<<<dim3(Jj / 128, Cc / 64, Bn), blk, shW, stream>>>(x, Wk, klin, Cc, Cc);
  cmix_gemm<<<dim3(Jj / 128, Cc / 64, Bn), blk, shW, stream>>>(x, Wv, vlin, Cc, Cc);

  // 2) FAVOR+ diagonal terms
  diag_kernel<<<dim3(BHn, Nn / 256), blk, 0, stream>>>(qlin, diagq);
  diag_kernel<<<dim3(BHn, Nn / 256), blk, 0, stream>>>(klin, diagk);

  // 3) feature-map GEMMs (query: fused rowmax+exp; key: raw dd)
  feat_gemm<<<dim3(Nn / 128, BHn), blk, shP, stream>>>(qlin, w_proj, diagq, qf, 1);
  feat_gemm<<<dim3(Nn / 128, BHn), blk, shP, stream>>>(klin, w_proj, nullptr, kf, 0);

  // 4) key global stabilizer, exp, row sums
  gmax_part<<<2048, blk, 0, stream>>>(kf, partb);
  gmax_fin<<<1, blk, 0, stream>>>(partb, kstab);
  finish_kf<<<(int)(FEAT / 256), blk, 0, stream>>>(kf, diagk, kstab);
  ksum_kernel<<<BHn * Mf / 256, blk, 0, stream>>>(kf, ksumb);

  // 5) linear attention
  vperm_kernel<<<(int)((size_t)BHn * Nn * 96 / 256), blk, 0, stream>>>(vlin, vp);
  ctx_gemm<<<dim3(6, Mf / 16, BHn), dim3(32), 0, stream>>>(kf, vp, ctxb);
  dinv_kernel<<<BHn * Nn / 8, blk, 0, stream>>>(qf, ksumb, dinvb);
  out_gemm<<<dim3(6, Nn / 128, BHn), blk, 0, stream>>>(qf, ctxb, dinvb, attn);

  // 6) attention layernorm + residual -> skip
  ln_res<<<dim3(Nn, Bn), blk, 0, stream>>>(attn, x, g_attn, b_attn, skip);

  // 7) VN-MLP
  cmix_gemm<<<dim3(Jj / 128, HIDn / 64, Bn), blk, shW, stream>>>(skip, W_mlp1, h1, HIDn, Cc);
  cmix_gemm<<<dim3(Jj / 128, HIDn / 64, Bn), blk, shW, stream>>>(h1, W_dir, dbuf, HIDn, HIDn);
  leaky_kernel<<<Bn * HIDn * Nn / 256, blk, 0, stream>>>(dbuf, h1);
  cmix_gemm<<<dim3(Jj / 128, Cc / 64, Bn), blk, shW, stream>>>(h1, W_mlp2, mlp2, Cc, HIDn);

  // 8) final layernorm + skip -> out
  ln_res<<<dim3(Nn, Bn), blk, 0, stream>>>(mlp2, skip, g_mlp, b_mlp, out);
}